// SSMALayer_34179349741737
// MI455X (gfx1250) — compile-verified
//
#include <hip/hip_runtime.h>
#include <hip/hip_bf16.h>

typedef float v2f __attribute__((ext_vector_type(2)));
typedef float v8f __attribute__((ext_vector_type(8)));

// D = A(16x4) * B(4x16) + C, full fp32 (CDNA5 V_WMMA_F32_16X16X4_F32)
#define WMMA_F32(a, b, c) \
  __builtin_amdgcn_wmma_f32_16x16x4_f32(false, (a), false, (b), (short)0, (c), false, false)

constexpr int BATCH = 8;
constexpr int SEQ   = 2048;
constexpr int DIM   = 1024;
constexpr int HID   = 4096;
constexpr int MEM   = 256;
constexpr int ROWS  = 32;    // rows of (B*S) per workgroup
constexpr int HC    = 512;   // H-chunk kept in LDS between the two GEMMs
constexpr int KB    = 32;    // K-block per prefetch window

__device__ __forceinline__ v8f v8f_zero() {
  v8f z;
#pragma unroll
  for (int i = 0; i < 8; ++i) z[i] = 0.0f;
  return z;
}

// ---------------------------------------------------------------------------
// Fused FFN: out = x + relu(x @ w1^T + b1) @ w2^T + b2
// One WG = 32 rows x full 1024 cols; h never leaves the WGP (LDS chunking).
// ---------------------------------------------------------------------------
__global__ __launch_bounds__(256) void ffn_kernel(
    const float* __restrict__ x,  const float* __restrict__ w1,
    const float* __restrict__ b1, const float* __restrict__ w2,
    const float* __restrict__ b2, float* __restrict__ out) {
  extern __shared__ float smem[];
  float* x_s = smem;               // 32 x 1024 fp32 = 128 KB
  float* h_s = smem + ROWS * DIM;  // 32 x 512  fp32 =  64 KB

  const int tid  = threadIdx.x;
  const int lane = tid & 31;
  const int wv   = tid >> 5;            // wave 0..7
  const int lr   = lane & 15;           // row/col index inside a 16-tile
  const int kh   = (lane >> 4) * 2;     // K sub-offset (0 or 2) per §7.12.2
  const int r0   = blockIdx.x * ROWS;   // global row base in flattened (B*S)

  // cooperative stage of the x tile (coalesced b128 loads)
  {
    const float4* src = reinterpret_cast<const float4*>(x + (size_t)r0 * DIM);
    float4*       dst = reinterpret_cast<float4*>(x_s);
    for (int i = tid; i < ROWS * DIM / 4; i += 256) dst[i] = src[i];
  }
  __syncthreads();

  // persistent output accumulators: wave owns 32 rows x 128 cols -> 16 C tiles
  v8f oacc[2][8];
#pragma unroll
  for (int mt = 0; mt < 2; ++mt)
#pragma unroll
    for (int nt = 0; nt < 8; ++nt) oacc[mt][nt] = v8f_zero();

  for (int hb = 0; hb < HID / HC; ++hb) {
    const int hcol0 = hb * HC;

    // ---- GEMM1: h_s[32][512] = relu(x_tile @ w1^T + b1), wave owns 64 cols
    v8f hacc[2][4];
#pragma unroll
    for (int mt = 0; mt < 2; ++mt)
#pragma unroll
      for (int nt = 0; nt < 4; ++nt) hacc[mt][nt] = v8f_zero();

    const int wc = wv * 64;
    const float* w1base[4];
#pragma unroll
    for (int nt = 0; nt < 4; ++nt)
      w1base[nt] = w1 + (size_t)(hcol0 + wc + nt * 16 + lr) * DIM + kh;

    for (int kb = 0; kb < DIM; kb += KB) {
      if (kb + KB < DIM) {  // pull next K-block toward the WGP cache
#pragma unroll
        for (int nt = 0; nt < 4; ++nt)
          __builtin_prefetch(w1base[nt] + kb + KB, 0, 3);
      }
#pragma unroll
      for (int kk = 0; kk < KB; kk += 4) {
        const int k = kb + kk;
        v2f a[2];
#pragma unroll
        for (int mt = 0; mt < 2; ++mt)
          a[mt] = *reinterpret_cast<const v2f*>(&x_s[(mt * 16 + lr) * DIM + k + kh]);
        v2f b[4];  // independent dests -> loads issue back-to-back
#pragma unroll
        for (int nt = 0; nt < 4; ++nt)
          b[nt] = *reinterpret_cast<const v2f*>(w1base[nt] + k);
#pragma unroll
        for (int nt = 0; nt < 4; ++nt)
#pragma unroll
          for (int mt = 0; mt < 2; ++mt)
            hacc[mt][nt] = WMMA_F32(a[mt], b[nt], hacc[mt][nt]);
      }
    }
    // bias + relu -> LDS (C layout: M = r + 8*lane[4], N = lane&15)
#pragma unroll
    for (int mt = 0; mt < 2; ++mt)
#pragma unroll
      for (int nt = 0; nt < 4; ++nt) {
        const int n_loc = wc + nt * 16 + lr;
        const float bias = b1[hcol0 + n_loc];
#pragma unroll
        for (int r = 0; r < 8; ++r) {
          const int m = mt * 16 + r + 8 * (lane >> 4);
          float v = hacc[mt][nt][r] + bias;
          h_s[m * HC + n_loc] = v > 0.0f ? v : 0.0f;
        }
      }
    __syncthreads();

    // ---- GEMM2 partial: oacc += h_chunk @ w2^T[:, hcol0:hcol0+HC]
    const int oc = wv * 128;
    const float* w2base[8];
#pragma unroll
    for (int nt = 0; nt < 8; ++nt)
      w2base[nt] = w2 + (size_t)(oc + nt * 16 + lr) * HID + hcol0 + kh;

    for (int kb = 0; kb < HC; kb += KB) {
      if (kb + KB < HC) {
#pragma unroll
        for (int nt = 0; nt < 8; ++nt)
          __builtin_prefetch(w2base[nt] + kb + KB, 0, 3);
      }
#pragma unroll
      for (int kk = 0; kk < KB; kk += 4) {
        const int k = kb + kk;
        v2f a[2];
#pragma unroll
        for (int mt = 0; mt < 2; ++mt)
          a[mt] = *reinterpret_cast<const v2f*>(&h_s[(mt * 16 + lr) * HC + k + kh]);
        v2f b[8];
#pragma unroll
        for (int nt = 0; nt < 8; ++nt)
          b[nt] = *reinterpret_cast<const v2f*>(w2base[nt] + k);
#pragma unroll
        for (int nt = 0; nt < 8; ++nt)
#pragma unroll
          for (int mt = 0; mt < 2; ++mt)
            oacc[mt][nt] = WMMA_F32(a[mt], b[nt], oacc[mt][nt]);
      }
    }
    __syncthreads();  // h_s reused next chunk
  }

  // epilogue: out = oacc + x + b2 (coalesced across lanes within a 16-tile)
#pragma unroll
  for (int mt = 0; mt < 2; ++mt)
#pragma unroll
    for (int nt = 0; nt < 8; ++nt) {
      const int n = wv * 128 + nt * 16 + lr;
      const float bias = b2[n];
#pragma unroll
      for (int r = 0; r < 8; ++r) {
        const int m = mt * 16 + r + 8 * (lane >> 4);
        out[(size_t)(r0 + m) * DIM + n] = oacc[mt][nt][r] + x_s[m * DIM + n] + bias;
      }
    }
}

// ---------------------------------------------------------------------------
// x_proj = x @ win_w^T + win_b  -> workspace (16384 x 256 fp32)
// ---------------------------------------------------------------------------
__global__ __launch_bounds__(256) void xproj_kernel(
    const float* __restrict__ x, const float* __restrict__ win_w,
    const float* __restrict__ win_b, float* __restrict__ xp) {
  extern __shared__ float smem[];
  float* x_s = smem;  // 32 x 1024 = 128 KB

  const int tid  = threadIdx.x;
  const int lane = tid & 31;
  const int wv   = tid >> 5;
  const int lr   = lane & 15;
  const int kh   = (lane >> 4) * 2;
  const int r0   = blockIdx.x * ROWS;

  {
    const float4* src = reinterpret_cast<const float4*>(x + (size_t)r0 * DIM);
    float4*       dst = reinterpret_cast<float4*>(x_s);
    for (int i = tid; i < ROWS * DIM / 4; i += 256) dst[i] = src[i];
  }
  __syncthreads();

  v8f acc[2][2];
#pragma unroll
  for (int mt = 0; mt < 2; ++mt)
#pragma unroll
    for (int nt = 0; nt < 2; ++nt) acc[mt][nt] = v8f_zero();

  const int wc = wv * 32;  // wave owns 32 of 256 output cols
  const float* wbase[2];
#pragma unroll
  for (int nt = 0; nt < 2; ++nt)
    wbase[nt] = win_w + (size_t)(wc + nt * 16 + lr) * DIM + kh;

  for (int kb = 0; kb < DIM; kb += KB) {
    if (kb + KB < DIM) {
#pragma unroll
      for (int nt = 0; nt < 2; ++nt)
        __builtin_prefetch(wbase[nt] + kb + KB, 0, 3);
    }
#pragma unroll
    for (int kk = 0; kk < KB; kk += 4) {
      const int k = kb + kk;
      v2f a[2];
#pragma unroll
      for (int mt = 0; mt < 2; ++mt)
        a[mt] = *reinterpret_cast<const v2f*>(&x_s[(mt * 16 + lr) * DIM + k + kh]);
      v2f b[2];
#pragma unroll
      for (int nt = 0; nt < 2; ++nt)
        b[nt] = *reinterpret_cast<const v2f*>(wbase[nt] + k);
#pragma unroll
      for (int nt = 0; nt < 2; ++nt)
#pragma unroll
        for (int mt = 0; mt < 2; ++mt)
          acc[mt][nt] = WMMA_F32(a[mt], b[nt], acc[mt][nt]);
    }
  }
#pragma unroll
  for (int mt = 0; mt < 2; ++mt)
#pragma unroll
    for (int nt = 0; nt < 2; ++nt) {
      const int n = wc + nt * 16 + lr;
      const float bias = win_b[n];
#pragma unroll
      for (int r = 0; r < 8; ++r) {
        const int m = mt * 16 + r + 8 * (lane >> 4);
        xp[(size_t)(r0 + m) * MEM + n] = acc[mt][nt][r] + bias;
      }
    }
}

// ---------------------------------------------------------------------------
// Sequential scan: one WG per batch chain. wstate transposed+packed in LDS
// (conflict-free [kb][j] float4 layout). Exact top-k threshold via bitwise
// radix-select on monotone float keys in wave 0 (3 barriers/step).
// ---------------------------------------------------------------------------
__global__ __launch_bounds__(256) void scan_kernel(
    const float* __restrict__ state, const float* __restrict__ wstate,
    const float* __restrict__ gamma_p, const int* __restrict__ topk_p,
    const float* __restrict__ xp, float* __restrict__ mem_out) {
  extern __shared__ float smem[];
  float4* wpk   = reinterpret_cast<float4*>(smem);  // [64][256] f4 = 256 KB
  float*  g_s   = smem + 64 * 256 * 4;              // 256
  float*  su_s  = g_s + MEM;                        // 256
  float*  thr_s = su_s + MEM;                       // 1

  const int tid = threadIdx.x;  // 0..255 == output index j
  const int b   = blockIdx.x;   // chain

  // stage wstate packed: wpk[kb*256 + j] = wstate[j][4kb .. 4kb+3]
  for (int i = tid; i < 64 * MEM; i += 256) {
    const int kb = i >> 8, j = i & 255;
    wpk[kb * 256 + j] = *reinterpret_cast<const float4*>(wstate + (size_t)j * MEM + kb * 4);
  }
  const float g0 = state[b * MEM + tid];
  g_s[tid] = g0;
  float memv = g0;
  const float gamma = *gamma_p;
  const int   K     = *topk_p;
  __syncthreads();

  for (int t = 0; t < SEQ; ++t) {
    // su_j = relu( sum_k g_k * wstate[j][k] + xp[b][t][j] )
    float su = xp[((size_t)b * SEQ + t) * MEM + tid];
#pragma unroll 4
    for (int kb = 0; kb < 64; ++kb) {
      const float4 w  = wpk[kb * 256 + tid];                             // conflict-free
      const float4 g4 = *reinterpret_cast<const float4*>(&g_s[kb * 4]);  // broadcast
      su += w.x * g4.x + w.y * g4.y + w.z * g4.z + w.w * g4.w;
    }
    su = su > 0.0f ? su : 0.0f;
    su_s[tid] = su;
    __syncthreads();

    if (tid < 32) {  // wave 0: exact K-th-largest of 256 values
      unsigned key[8];
#pragma unroll
      for (int i = 0; i < 8; ++i) {
        const unsigned u = __float_as_uint(su_s[tid + 32 * i]);
        key[i] = (u & 0x80000000u) ? ~u : (u | 0x80000000u);  // monotone map
      }
      unsigned thr_key = 0u;
      for (int bit = 31; bit >= 0; --bit) {
        const unsigned cand = thr_key | (1u << bit);
        int c = 0;
#pragma unroll
        for (int i = 0; i < 8; ++i) c += (key[i] >= cand) ? 1 : 0;
#pragma unroll
        for (int off = 16; off > 0; off >>= 1) c += __shfl_xor(c, off, 32);
        if (c >= K) thr_key = cand;
      }
      if (tid == 0) {
        const unsigned u =
            (thr_key & 0x80000000u) ? (thr_key & 0x7fffffffu) : ~thr_key;
        thr_s[0] = __uint_as_float(u);
      }
    }
    __syncthreads();

    const float thr = thr_s[0];
    const float g   = su * (su >= thr ? 1.0f : 0.0f);
    memv            = gamma * memv + (1.0f - gamma) * g;
    g_s[tid]        = g;
    __syncthreads();
  }
  mem_out[b * MEM + tid] = memv;
}

// ---------------------------------------------------------------------------
extern "C" void kernel_launch(void* const* d_in, const int* in_sizes, int n_in,
                              void* d_out, int out_size, void* d_ws, size_t ws_size,
                              hipStream_t stream) {
  (void)in_sizes; (void)n_in; (void)out_size; (void)ws_size;
  const float* x      = (const float*)d_in[0];
  const float* state  = (const float*)d_in[1];
  // d_in[2] = U_w, d_in[3] = V_w : unused by the reference
  const float* w1     = (const float*)d_in[4];
  const float* b1     = (const float*)d_in[5];
  const float* w2     = (const float*)d_in[6];
  const float* b2     = (const float*)d_in[7];
  const float* win_w  = (const float*)d_in[8];
  const float* win_b  = (const float*)d_in[9];
  const float* wstate = (const float*)d_in[10];
  const float* gamma  = (const float*)d_in[11];
  const int*   topk   = (const int*)d_in[12];

  float* out     = (float*)d_out;
  float* mem_out = out + (size_t)BATCH * SEQ * DIM;  // memory follows outputs
  float* xp      = (float*)d_ws;                     // 16384 x 256 fp32 = 16 MB

  const int row_tiles = BATCH * SEQ / ROWS;  // 512
  const size_t ffn_lds   = (size_t)(ROWS * DIM + ROWS * HC) * sizeof(float);  // 192 KB
  const size_t xproj_lds = (size_t)(ROWS * DIM) * sizeof(float);              // 128 KB
  const size_t scan_lds  = (size_t)(64 * MEM * 4 + MEM + MEM + 4) * sizeof(float);

  ffn_kernel<<<row_tiles, 256, ffn_lds, stream>>>(x, w1, b1, w2, b2, out);
  xproj_kernel<<<row_tiles, 256, xproj_lds, stream>>>(x, win_w, win_b, xp);
  scan_kernel<<<BATCH, 256, scan_lds, stream>>>(state, wstate, gamma, topk, xp, mem_out);
}